// SparseCorrBlock_8607114461669
// MI455X (gfx1250) — compile-verified
//
#include <hip/hip_runtime.h>
#include <math.h>
#include <stdint.h>

// ---------------------------------------------------------------------------
// Problem constants (from reference): B=2, M=4096, C=256, H=W=64
// corr: (B*M, 64, 64) fp32 = 128 MiB scratch in d_ws
// out : (B, 4*81=324, M) fp32
// ---------------------------------------------------------------------------
#define M_TOT 4096
#define N_TOT 4096   // H*W
#define K_TOT 256
#define B_TOT 2

#define MB 128       // block tile M
#define NB 64        // block tile N
#define KB 32        // K chunk staged in LDS
#define AP 36        // padded LDS pitch for A chunk (144 B, 16B aligned)
#define NCHUNK (K_TOT / KB)   // 8

typedef __attribute__((ext_vector_type(2))) float v2f;
typedef __attribute__((ext_vector_type(8))) float v8f;

// ---------------------------------------------------------------------------
// gfx1250 async global->LDS path (ASYNCcnt). Builtin params are
// (global v4i*, lds v4i*, imm offset, imm cpol).
// ---------------------------------------------------------------------------
#if defined(__has_builtin)
#if __has_builtin(__builtin_amdgcn_global_load_async_to_lds_b128)
#define HAVE_ASYNC_LDS 1
#endif
#endif
#ifndef HAVE_ASYNC_LDS
#define HAVE_ASYNC_LDS 0
#endif

#if defined(__has_builtin)
#if __has_builtin(__builtin_amdgcn_s_wait_asynccnt)
#define WAIT_ASYNC(n) __builtin_amdgcn_s_wait_asynccnt(n)
#endif
#endif
#ifndef WAIT_ASYNC
#define WAIT_ASYNC(n) asm volatile("s_wait_asynccnt %0" ::"n"(n) : "memory")
#endif

#if HAVE_ASYNC_LDS
typedef int v4i __attribute__((vector_size(16)));
typedef __attribute__((address_space(1))) v4i* gptr_v4i;
typedef __attribute__((address_space(3))) v4i* lptr_v4i;

__device__ __forceinline__ void async_b128(const float* g, float* l) {
  // generic->AS(1): numeric identity; generic->AS(3): LDS addr = addr[31:0]
  gptr_v4i gp = (gptr_v4i)(uintptr_t)g;
  lptr_v4i lp = (lptr_v4i)(uint32_t)(uintptr_t)l;
  __builtin_amdgcn_global_load_async_to_lds_b128(gp, lp, 0, 0);
}
#endif

// ---------------------------------------------------------------------------
// Kernel 1: corr[b,m,n] = sum_c fmap1[b,m,c] * fmap2[b,c,n]
// fp32 WMMA 16x16x4. Block: 256 threads = 8 waves; each wave computes a
// 16(M) x 64(N) strip as 4 accumulators of 16x16.
//  - A chunks: async global->LDS, double-buffered, s_wait_asynccnt pipeline.
//  - B chunks: global->VGPR->LDS stored K-pair interleaved (BsI[kp][n][2]) so
//    every WMMA B fragment is a single aligned ds_load_b64 (no repack movs).
// ---------------------------------------------------------------------------
__global__ __launch_bounds__(256) void sparse_corr_gemm_f32(
    const float* __restrict__ fmap1,   // (B, 4096, 256) row-major
    const float* __restrict__ fmap2,   // (B, 256, 4096) row-major
    float* __restrict__ corr)          // (B, 4096, 4096)
{
  __shared__ float As[2][MB * AP];              // 2 x 18 KB
  __shared__ float BsI[2][(KB / 2) * NB * 2];   // 2 x  8 KB, [kp][n][k&1]

  const int tid  = threadIdx.x;
  const int lane = tid & 31;
  const int wv   = tid >> 5;      // wave 0..7 -> M sub-tile
  const int l    = lane & 15;
  const int half = lane >> 4;     // K-pair select / M+8 select

  const int n0 = blockIdx.x * NB;
  const int m0 = blockIdx.y * MB;
  const int b  = blockIdx.z;

  const float* Ab = fmap1 + (size_t)b * M_TOT * K_TOT;
  const float* Bb = fmap2 + (size_t)b * K_TOT * N_TOT;

  // B staging coordinates for this thread: one K-pair row, one 4-wide N group
  const int bkp  = tid >> 4;      // 0..15  -> K rows {2bkp, 2bkp+1}
  const int bgrp = tid & 15;      // 0..15  -> N cols [4*bgrp, 4*bgrp+4)

  v8f acc[4];
  #pragma unroll
  for (int t = 0; t < 4; ++t) {
    v8f z = {0.f, 0.f, 0.f, 0.f, 0.f, 0.f, 0.f, 0.f};
    acc[t] = z;
  }

  auto stageA = [&](int kc, int buf) {
    #pragma unroll
    for (int i = 0; i < 4; ++i) {
      int u = tid + i * 256;          // 0..1023
      int row = u >> 3;               // 8 float4 groups per 32-float A row
      int grp = u & 7;
      const float* g = Ab + (size_t)(m0 + row) * K_TOT + kc + grp * 4;
      float* ldst = &As[buf][row * AP + grp * 4];
#if HAVE_ASYNC_LDS
      async_b128(g, ldst);
#else
      *(float4*)ldst = *(const float4*)g;
#endif
    }
  };

  auto loadB = [&](int kc, float4& r0, float4& r1) {
    const float* g0 = Bb + (size_t)(kc + 2 * bkp) * N_TOT + n0 + bgrp * 4;
    r0 = *(const float4*)g0;
    r1 = *(const float4*)(g0 + N_TOT);
  };

  auto storeB = [&](int buf, float4 r0, float4 r1) {
    float* d = &BsI[buf][(bkp * NB + bgrp * 4) * 2];
    float4 s0 = {r0.x, r1.x, r0.y, r1.y};   // n, n+1 interleaved pairs
    float4 s1 = {r0.z, r1.z, r0.w, r1.w};   // n+2, n+3
    *(float4*)(d)     = s0;
    *(float4*)(d + 4) = s1;
  };

  auto compute = [&](int buf) {
    const float* AsC = As[buf];
    const float* BsC = BsI[buf];
    #pragma unroll
    for (int kk = 0; kk < KB; kk += 4) {
      // A 16x4 fragment: lanes 0-15 K={kk,kk+1}, lanes 16-31 K={kk+2,kk+3}
      v2f afrag = *(const v2f*)(&AsC[(wv * 16 + l) * AP + kk + 2 * half]);
      const int kp = (kk >> 1) + half;      // pair index (kb = kk + 2*half)
      #pragma unroll
      for (int t = 0; t < 4; ++t) {
        // contiguous b64 fragment: {B[kb][n], B[kb+1][n]}
        v2f bfrag = *(const v2f*)(&BsC[(kp * NB + t * 16 + l) * 2]);
        acc[t] = __builtin_amdgcn_wmma_f32_16x16x4_f32(
            false, afrag, false, bfrag, (short)0, acc[t], false, false);
      }
    }
  };

#if HAVE_ASYNC_LDS
  // Software pipeline: chunk i+1 (A async, B global loads) overlaps WMMA on i.
  float4 rb0, rb1;
  stageA(0, 0);
  loadB(0, rb0, rb1);
  storeB(0, rb0, rb1);
  for (int ci = 0; ci < NCHUNK; ++ci) {
    const int cur = ci & 1;
    const int nxt = cur ^ 1;
    if (ci + 1 < NCHUNK) {
      stageA((ci + 1) * KB, nxt);
      loadB((ci + 1) * KB, rb0, rb1);
      WAIT_ASYNC(4);                 // oldest 4 (chunk ci A) done, in-order
    } else {
      WAIT_ASYNC(0);
    }
    __syncthreads();                  // chunk ci fully visible to all waves
    compute(cur);
    if (ci + 1 < NCHUNK) storeB(nxt, rb0, rb1);
    __syncthreads();                  // buf cur free before chunk ci+2 staged
  }
#else
  for (int ci = 0; ci < NCHUNK; ++ci) {
    float4 rb0, rb1;
    stageA(ci * KB, 0);
    loadB(ci * KB, rb0, rb1);
    storeB(0, rb0, rb1);
    __syncthreads();
    compute(0);
    __syncthreads();
  }
#endif

  // ---- epilogue: C/D layout -> row = 8*half + r, col = l (per 16x16 tile)
  float* Cb = corr + (size_t)b * M_TOT * N_TOT;
  #pragma unroll
  for (int t = 0; t < 4; ++t) {
    #pragma unroll
    for (int r = 0; r < 8; ++r) {
      int row = m0 + wv * 16 + half * 8 + r;
      int col = n0 + t * 16 + l;
      Cb[(size_t)row * N_TOT + col] = acc[t][r];
    }
  }
}

// ---------------------------------------------------------------------------
// Kernel 2: per (b,m): build 4-level pyramid of the 64x64 corr slice in LDS,
// bilinear-sample 9x9 windows (zeros padding) at c/2^lvl, write (B,324,M).
// ---------------------------------------------------------------------------
#define L0_OFF 0       // 64x64 = 4096
#define L1_OFF 4096    // 32x32 = 1024
#define L2_OFF 5120    // 16x16 = 256
#define L3_OFF 5376    // 8x8   = 64
#define L_TOT  5440

__global__ __launch_bounds__(256) void sparse_corr_pyr_sample(
    const float* __restrict__ corr,    // (B*M, 4096)
    const float* __restrict__ cents,   // (B, M, 2)  [x, y]
    float* __restrict__ out)           // (B, 324, M)
{
  __shared__ float L[L_TOT];

  const int tid = threadIdx.x;
  const int bm  = blockIdx.x;          // 0..8191
  const int b   = bm >> 12;
  const int m   = bm & (M_TOT - 1);

  // ---- 16 KB corr row into LDS
  const float* rowp = corr + (size_t)bm * N_TOT;
  #pragma unroll
  for (int i = 0; i < 4; ++i) {
    int u = tid + i * 256;             // 0..1023
#if HAVE_ASYNC_LDS
    async_b128(rowp + u * 4, &L[L0_OFF + u * 4]);
#else
    *(float4*)(&L[L0_OFF + u * 4]) = *(const float4*)(rowp + u * 4);
#endif
  }
#if HAVE_ASYNC_LDS
  WAIT_ASYNC(0);
#endif
  __syncthreads();

  // ---- level 1: 32x32 avg-pool of 64x64
  for (int i = tid; i < 1024; i += 256) {
    int y = i >> 5, x = i & 31;
    const float* p = &L[L0_OFF + (2 * y) * 64 + 2 * x];
    L[L1_OFF + i] = 0.25f * (p[0] + p[1] + p[64] + p[65]);
  }
  __syncthreads();

  // ---- level 2: 16x16
  if (tid < 256) {
    int y = tid >> 4, x = tid & 15;
    const float* p = &L[L1_OFF + (2 * y) * 32 + 2 * x];
    L[L2_OFF + tid] = 0.25f * (p[0] + p[1] + p[32] + p[33]);
  }
  __syncthreads();

  // ---- level 3: 8x8
  if (tid < 64) {
    int y = tid >> 3, x = tid & 7;
    const float* p = &L[L2_OFF + (2 * y) * 16 + 2 * x];
    L[L3_OFF + tid] = 0.25f * (p[0] + p[1] + p[16] + p[17]);
  }
  __syncthreads();

  const float cx = cents[(size_t)bm * 2 + 0];
  const float cy = cents[(size_t)bm * 2 + 1];

  // ---- 324 samples: k = lvl*81 + i*9 + j ; x = cx/2^lvl + (i-4), y = cy/2^lvl + (j-4)
  for (int k = tid; k < 324; k += 256) {
    int lvl = k / 81;
    int r   = k - lvl * 81;
    int di  = r / 9 - 4;               // x offset (first axis of the 9x9)
    int dj  = r - (r / 9) * 9 - 4;     // y offset

    int Wl; int off;
    if (lvl == 0)      { Wl = 64; off = L0_OFF; }
    else if (lvl == 1) { Wl = 32; off = L1_OFF; }
    else if (lvl == 2) { Wl = 16; off = L2_OFF; }
    else               { Wl = 8;  off = L3_OFF; }

    float inv = 1.0f / (float)(1 << lvl);
    float x = cx * inv + (float)di;
    float y = cy * inv + (float)dj;

    float x0f = floorf(x), y0f = floorf(y);
    int   x0  = (int)x0f,  y0  = (int)y0f;
    float wx1 = x - x0f, wx0 = 1.0f - wx1;
    float wy1 = y - y0f, wy0 = 1.0f - wy1;

    const float* base = &L[off];
    float v00 = 0.f, v10 = 0.f, v01 = 0.f, v11 = 0.f;
    bool xa = (x0 >= 0) && (x0 <= Wl - 1);
    bool xb = (x0 + 1 >= 0) && (x0 + 1 <= Wl - 1);
    bool ya = (y0 >= 0) && (y0 <= Wl - 1);
    bool yb = (y0 + 1 >= 0) && (y0 + 1 <= Wl - 1);
    if (xa && ya) v00 = base[y0 * Wl + x0];
    if (xb && ya) v10 = base[y0 * Wl + x0 + 1];
    if (xa && yb) v01 = base[(y0 + 1) * Wl + x0];
    if (xb && yb) v11 = base[(y0 + 1) * Wl + x0 + 1];

    float val = wy0 * wx0 * v00 + wy0 * wx1 * v10
              + wy1 * wx0 * v01 + wy1 * wx1 * v11;

    out[((size_t)b * 324 + k) * M_TOT + m] = val;
  }
}

// ---------------------------------------------------------------------------
extern "C" void kernel_launch(void* const* d_in, const int* in_sizes, int n_in,
                              void* d_out, int out_size, void* d_ws, size_t ws_size,
                              hipStream_t stream) {
  const float* fmap1 = (const float*)d_in[0];  // (2, 4096, 256)
  const float* fmap2 = (const float*)d_in[1];  // (2, 256, 64, 64)
  const float* cents = (const float*)d_in[2];  // (2, 4096, 2)
  float* corr = (float*)d_ws;                  // needs 2*4096*4096*4 = 128 MiB
  float* out  = (float*)d_out;                 // (2, 324, 4096)

  dim3 g1(N_TOT / NB, M_TOT / MB, B_TOT);      // (64, 32, 2)
  sparse_corr_gemm_f32<<<g1, 256, 0, stream>>>(fmap1, fmap2, corr);

  sparse_corr_pyr_sample<<<B_TOT * M_TOT, 256, 0, stream>>>(corr, cents, out);
}